// Upsample_46110768890055
// MI455X (gfx1250) — compile-verified
//
#include <hip/hip_runtime.h>

typedef float v4f __attribute__((ext_vector_type(4)));
typedef int   v4i __attribute__((ext_vector_type(4)));

// Pointee-typed, address-space-qualified pointers matching the builtin's
// signature: (int4 AS1* gsrc, int4 AS3* ldst, imm offset, imm cpol).
typedef v4i __attribute__((address_space(1))) as1_v4i;   // global int4
typedef v4i __attribute__((address_space(3))) as3_v4i;   // LDS int4

#if defined(__HIP_DEVICE_COMPILE__) &&                                   \
    __has_builtin(__builtin_amdgcn_global_load_async_to_lds_b128) &&     \
    __has_builtin(__builtin_amdgcn_s_wait_asynccnt)
#define USE_ASYNC_LDS 1
#else
#define USE_ASYNC_LDS 0
#endif

#define TILE_H 64
#define TILE_W 64

// One workgroup per (n,c) plane. 64x64 fp32 plane = 16 KB in LDS.
// Polyphase 2x upsample with 4-tap separable kernel:
//   even out rows: u3*x[m-1] + u1*x[m] ; odd: u2*x[m] + u0*x[m+1]  (same in x)
__global__ __launch_bounds__(256) void upfirdn2d_up2_k4(
    const float* __restrict__ x,
    const float* __restrict__ kern,   // 4x4 separable kernel, row-major
    float* __restrict__ out)
{
    __shared__ __align__(16) float tile[TILE_H * TILE_W];
    __shared__ float ku[4];

    const int tid = threadIdx.x;
    const int p   = blockIdx.x;
    const float* __restrict__ src = x + (size_t)p * (TILE_H * TILE_W);

    // ---- Stage input plane into LDS ------------------------------------
#if USE_ASYNC_LDS
    for (int i = tid; i < (TILE_H * TILE_W) / 4; i += 256) {
        void* gp = (void*)(src + 4 * i);           // drop const
        __builtin_amdgcn_global_load_async_to_lds_b128(
            (as1_v4i*)gp,
            (as3_v4i*)(tile + 4 * i),
            /*offset=*/0, /*cpol=*/0);
    }
#else
    for (int i = tid; i < (TILE_H * TILE_W) / 4; i += 256)
        ((v4f*)tile)[i] = ((const v4f*)src)[i];
#endif

    // 1-D kernel = row sums of the (normalized) separable 2-D kernel.
    if (tid < 4)
        ku[tid] = kern[4*tid+0] + kern[4*tid+1] + kern[4*tid+2] + kern[4*tid+3];

#if USE_ASYNC_LDS
    __builtin_amdgcn_s_wait_asynccnt(0);
#endif
    __syncthreads();

    const float u0 = ku[0], u1 = ku[1], u2 = ku[2], u3 = ku[3];
    float* __restrict__ dst = out + (size_t)p * (4 * TILE_H * TILE_W);

    // 128 output rows x 32 float4-quads; each 32-lane wave handles one row
    // per iteration (row wave-uniform, q = lane id) -> 512B coalesced stores.
    for (int i = tid; i < (2*TILE_H) * (2*TILE_W) / 4; i += 256) {
        const int row = i >> 5;       // 0..127
        const int q   = i & 31;       // float4 index along output x
        const int my  = row >> 1;

        int rA, rB; float wA, wB;
        if ((row & 1) == 0) { rA = my - 1; rB = my;     wA = u3; wB = u1; }
        else                { rA = my;     rB = my + 1; wA = u2; wB = u0; }
        if (rA < 0)          { rA = 0;          wA = 0.0f; }   // top edge
        if (rB > TILE_H - 1) { rB = TILE_H - 1; wB = 0.0f; }   // bottom edge

        const float* tA = tile + rA * TILE_W;
        const float* tB = tile + rB * TILE_W;

        // Vertical blend for input columns 2q-1 .. 2q+2 (zero outside plane)
        float v[4];
#pragma unroll
        for (int j = 0; j < 4; ++j) {
            int c  = 2*q - 1 + j;
            int cc = c < 0 ? 0 : (c > TILE_W - 1 ? TILE_W - 1 : c);
            float m = (c == cc) ? 1.0f : 0.0f;
            v[j] = (wA * tA[cc] + wB * tB[cc]) * m;
        }

        // Horizontal polyphase blend -> 4 consecutive output pixels
        v4f o;
        o.x = u3 * v[0] + u1 * v[1];   // ox = 4q   (even phase)
        o.y = u2 * v[1] + u0 * v[2];   // ox = 4q+1 (odd phase)
        o.z = u3 * v[1] + u1 * v[2];   // ox = 4q+2
        o.w = u2 * v[2] + u0 * v[3];   // ox = 4q+3

        // Output stream (537 MB) >> L2 (192 MB): non-temporal store.
        __builtin_nontemporal_store(o, (v4f*)(dst + row * (2*TILE_W)) + q);
    }
}

extern "C" void kernel_launch(void* const* d_in, const int* in_sizes, int n_in,
                              void* d_out, int out_size, void* d_ws, size_t ws_size,
                              hipStream_t stream) {
    const float* x = (const float*)d_in[0];
    const float* k = (const float*)d_in[1];
    float* out     = (float*)d_out;
    const int planes = in_sizes[0] / (TILE_H * TILE_W);   // N*C = 8192
    upfirdn2d_up2_k4<<<planes, 256, 0, stream>>>(x, k, out);
    (void)n_in; (void)out_size; (void)d_ws; (void)ws_size;
}